// guideEncoder_body_4913442586838
// MI455X (gfx1250) — compile-verified
//
#include <hip/hip_runtime.h>

typedef __attribute__((ext_vector_type(2))) float v2f;
typedef __attribute__((ext_vector_type(8))) float v8f;

#define GAT_F 128      // feature width (both layers): NFEAT = HEADS*HID = 128
#define GAT_H 4        // heads
#define GAT_C 32       // per-head channels
#define GAT_SLOPE 0.2f

// ---- order-preserving monotone transform for float atomicMax via u32 ----
__device__ __forceinline__ unsigned f2mono(float x) {
  unsigned u = __float_as_uint(x);
  return (u & 0x80000000u) ? ~u : (u | 0x80000000u);
}
__device__ __forceinline__ float mono2f(unsigned u) {
  return (u & 0x80000000u) ? __uint_as_float(u & 0x7fffffffu)
                           : __uint_as_float(~u);
}

// =====================================================================
// GEMM: Y[N,128] = X[N,128] @ W[128,128], f32 WMMA 16x16x4.
// One wave computes a 16x128 output stripe (8 column tiles of 16).
// W is staged into LDS with (k-pair, n) interleave so each B fragment
// is a single aligned 8-byte LDS load.
// =====================================================================
__global__ __launch_bounds__(256) void gat_gemm_wmma(const float* __restrict__ X,
                                                     const float* __restrict__ W,
                                                     float* __restrict__ Y,
                                                     int ntiles) {
  __shared__ float ldsW[GAT_F * GAT_F];  // 64 KB
  // cooperative stage: ldsW[((k>>1)*128 + n)*2 + (k&1)] = W[k*128+n]
  for (int idx = threadIdx.x; idx < GAT_F * GAT_F; idx += 256) {
    int k = idx >> 7;
    int n = idx & 127;
    ldsW[(((k >> 1) * GAT_F + n) << 1) + (k & 1)] = W[idx];
  }
  __syncthreads();

  int wave = blockIdx.x * (blockDim.x >> 5) + (threadIdx.x >> 5);
  if (wave >= ntiles) return;          // whole wave exits: EXEC stays all-ones
  int lane = threadIdx.x & 31;
  int m  = lane & 15;                  // row within tile (also column within ctile)
  int kh = lane >> 4;                  // K half-select (0 -> K0,K1 ; 1 -> K2,K3)

  const float* xrow = X + (size_t)(wave * 16 + m) * GAT_F;

  v8f acc[8] = {};                     // 8 x (16x16) f32 accumulators
  for (int kk = 0; kk < GAT_F; kk += 4) {
    int k0 = kk + (kh << 1);           // even
    v2f a;
    a.x = xrow[k0];
    a.y = xrow[k0 + 1];
#pragma unroll
    for (int t = 0; t < 8; ++t) {
      int n = (t << 4) + m;
      v2f b = *(const v2f*)&ldsW[(((k0 >> 1) * GAT_F + n) << 1)];
      acc[t] = __builtin_amdgcn_wmma_f32_16x16x4_f32(
          false, a, false, b, (short)0, acc[t], false, false);
    }
  }

#pragma unroll
  for (int t = 0; t < 8; ++t) {
    int n = (t << 4) + m;
#pragma unroll
    for (int r = 0; r < 8; ++r) {
      int row = wave * 16 + r + (kh << 3);   // VGPR r holds M=r (lanes<16) / r+8
      Y[(size_t)row * GAT_F + n] = acc[t][r];
    }
  }
}

// =====================================================================
// a_l[n,h] = <h[n,h,:], att_l[h,:]> ; a_r likewise. Also init M (=-inf
// in monotone space => 0u) and S (=0).
// =====================================================================
__global__ void gat_attdot_init(const float* __restrict__ Hm,
                                const float* __restrict__ attl,
                                const float* __restrict__ attr,
                                float* __restrict__ AL, float* __restrict__ AR,
                                unsigned* __restrict__ M, float* __restrict__ S,
                                int total) {
  int idx = blockIdx.x * blockDim.x + threadIdx.x;
  if (idx >= total) return;
  int n = idx >> 2;
  int h = idx & 3;
  const float* hp = Hm + (size_t)n * GAT_F + h * GAT_C;
  const float* al = attl + h * GAT_C;
  const float* ar = attr + h * GAT_C;
  float sl = 0.f, sr = 0.f;
#pragma unroll 8
  for (int c = 0; c < GAT_C; ++c) {
    float v = hp[c];
    sl = fmaf(v, al[c], sl);
    sr = fmaf(v, ar[c], sr);
  }
  AL[idx] = sl;
  AR[idx] = sr;
  M[idx]  = 0u;    // monotone(-inf)
  S[idx]  = 0.f;
}

// =====================================================================
// per-(edge,head): logit = leaky_relu(a_l[src]+a_r[dst]) * ew ;
// store logit, atomic segment-max over dst.
// =====================================================================
__global__ void gat_edge_logit_max(const int* __restrict__ src,
                                   const int* __restrict__ dst,
                                   const float* __restrict__ ew,
                                   const float* __restrict__ AL,
                                   const float* __restrict__ AR,
                                   float* __restrict__ LOGB,
                                   unsigned* __restrict__ M, int total) {
  int idx = blockIdx.x * blockDim.x + threadIdx.x;
  if (idx >= total) return;
  int e = idx >> 2;
  int h = idx & 3;
  int s = src[e];
  int d = dst[e];
  float v = AL[s * 4 + h] + AR[d * 4 + h];
  v = (v > 0.f) ? v : GAT_SLOPE * v;
  v *= ew[e];
  LOGB[idx] = v;
  atomicMax(&M[d * 4 + h], f2mono(v));
}

// =====================================================================
// per-(edge,head): ex = exp(logit - m[dst]) ; atomic segment-sum.
// =====================================================================
__global__ void gat_edge_exp_sum(const int* __restrict__ dst,
                                 float* __restrict__ LOGB,
                                 const unsigned* __restrict__ M,
                                 float* __restrict__ S, int total) {
  int idx = blockIdx.x * blockDim.x + threadIdx.x;
  if (idx >= total) return;
  int e = idx >> 2;
  int h = idx & 3;
  int d = dst[e];
  float ex = __expf(LOGB[idx] - mono2f(M[d * 4 + h]));
  LOGB[idx] = ex;
  atomicAdd(&S[d * 4 + h], ex);
}

// out[n,:] = bias broadcast
__global__ void gat_init_out(float* __restrict__ OUT,
                             const float* __restrict__ bias, int total) {
  int idx = blockIdx.x * blockDim.x + threadIdx.x;
  if (idx >= total) return;
  OUT[idx] = bias[idx & (GAT_F - 1)];
}

// =====================================================================
// weighted scatter-add: OUT[dst, h*32+c] += H[src, h*32+c] * alpha(e,h)
// two edges per 256-thread block; lane j covers channel j&127.
// =====================================================================
__global__ __launch_bounds__(256) void gat_scatter(const int* __restrict__ src,
                                                   const int* __restrict__ dst,
                                                   const float* __restrict__ Hm,
                                                   const float* __restrict__ ALPHA,
                                                   const float* __restrict__ S,
                                                   float* __restrict__ OUT, int E) {
  int e = blockIdx.x * 2 + (threadIdx.x >> 7);
  if (e >= E) return;
  int j = threadIdx.x & 127;
  int h = j >> 5;
  int s = src[e];
  int d = dst[e];
  float alpha = ALPHA[e * 4 + h] / (S[d * 4 + h] + 1e-16f);
  atomicAdd(&OUT[(size_t)d * GAT_F + j], Hm[(size_t)s * GAT_F + j] * alpha);
}

// =====================================================================
// Host-side layer driver (stream-ordered, graph-capture safe)
// =====================================================================
static void run_gat_layer(const float* xin, const float* W, const float* attl,
                          const float* attr, const float* bias, const int* src,
                          const int* dst, const float* ew, int N, int E,
                          float* Hbuf, float* AL, float* AR, float* S,
                          unsigned* M, float* LOGB, float* OUT,
                          hipStream_t stream) {
  int ntiles = N / 16;
  gat_gemm_wmma<<<(ntiles + 7) / 8, 256, 0, stream>>>(xin, W, Hbuf, ntiles);

  int nh = N * GAT_H;
  gat_attdot_init<<<(nh + 255) / 256, 256, 0, stream>>>(Hbuf, attl, attr, AL,
                                                        AR, M, S, nh);
  int eh = E * GAT_H;
  gat_edge_logit_max<<<(eh + 255) / 256, 256, 0, stream>>>(src, dst, ew, AL,
                                                           AR, LOGB, M, eh);
  gat_edge_exp_sum<<<(eh + 255) / 256, 256, 0, stream>>>(dst, LOGB, M, S, eh);

  int nf = N * GAT_F;
  gat_init_out<<<(nf + 255) / 256, 256, 0, stream>>>(OUT, bias, nf);
  gat_scatter<<<(E + 1) / 2, 256, 0, stream>>>(src, dst, Hbuf, LOGB, S, OUT, E);
}

extern "C" void kernel_launch(void* const* d_in, const int* in_sizes, int n_in,
                              void* d_out, int out_size, void* d_ws,
                              size_t ws_size, hipStream_t stream) {
  (void)n_in; (void)out_size; (void)ws_size;
  const float* x    = (const float*)d_in[0];
  const int*   eidx = (const int*)d_in[1];
  const float* ew   = (const float*)d_in[2];
  const float* W1   = (const float*)d_in[3];
  const float* al1  = (const float*)d_in[4];
  const float* ar1  = (const float*)d_in[5];
  const float* b1   = (const float*)d_in[6];
  const float* W2   = (const float*)d_in[7];
  const float* al2  = (const float*)d_in[8];
  const float* ar2  = (const float*)d_in[9];
  const float* b2   = (const float*)d_in[10];

  const int N = in_sizes[0] / GAT_F;   // 50000
  const int E = in_sizes[1] / 2;       // 800000
  const int* src = eidx;
  const int* dst = eidx + E;

  // workspace layout (floats): Hbuf | AGG | AL | AR | S | M(u32) | LOGB
  float* ws   = (float*)d_ws;
  float* Hbuf = ws;                         // N*128
  float* AGG  = Hbuf + (size_t)N * GAT_F;   // N*128 (layer-1 output)
  float* AL   = AGG + (size_t)N * GAT_F;    // N*4
  float* AR   = AL + (size_t)N * GAT_H;     // N*4
  float* S    = AR + (size_t)N * GAT_H;     // N*4
  unsigned* M = (unsigned*)(S + (size_t)N * GAT_H);  // N*4 (u32)
  float* LOGB = (float*)(M + (size_t)N * GAT_H);     // E*4

  float* out = (float*)d_out;

  run_gat_layer(x,   W1, al1, ar1, b1, src, dst, ew, N, E, Hbuf, AL, AR, S, M,
                LOGB, AGG, stream);
  run_gat_layer(AGG, W2, al2, ar2, b2, src, dst, ew, N, E, Hbuf, AL, AR, S, M,
                LOGB, out, stream);
}